// PolicyNetworkOptimized_72584947302716
// MI455X (gfx1250) — compile-verified
//
#include <hip/hip_runtime.h>
#include <math.h>

// ---------------------------------------------------------------------------
// PolicyNetwork scorer for MI455X (gfx1250, wave32).
//
// Factorization: disp_in @ W1d splits into three K=128 GEMMs (agv/order/
// waiting blocks of W1d), pickup into two. Heavy 25-GFLOP einsum collapses to
// ~0.13 GFLOP combine. GEMMs run on the matrix pipe with
// V_WMMA_F32_16X16X4_F32 (full f32, matches reference precision).
// ---------------------------------------------------------------------------

typedef float v2f __attribute__((ext_vector_type(2)));
typedef float v8f __attribute__((ext_vector_type(8)));

#define HAVE_WMMA_F32_16X16X4 0
#if defined(__has_builtin)
#if __has_builtin(__builtin_amdgcn_wmma_f32_16x16x4_f32)
#undef HAVE_WMMA_F32_16X16X4
#define HAVE_WMMA_F32_16X16X4 1
#endif
#endif

namespace pn {
constexpr int O = 100, M = 40, A = 64, D = 128;
constexpr int OM = O * M;                 // 4000
constexpr int LOGITS_PER_A = OM + M;      // 4040

// Workspace layout (floats), rows padded to multiples of 16 for WMMA tiles.
constexpr int UA_R = 64, UO_R = 112, UM_R = 48, PA_R = 64, PB_R = 48;
constexpr int OFF_UA = 0;
constexpr int OFF_UO = OFF_UA + UA_R * D;
constexpr int OFF_UM = OFF_UO + UO_R * D;
constexpr int OFF_PA = OFF_UM + UM_R * D;
constexpr int OFF_PB = OFF_PA + PA_R * D;
// total = 336*128 floats = 172,032 bytes of d_ws

constexpr int LSTR = 132;  // LDS row stride (floats): 16B-aligned, kills bank conflicts
}  // namespace pn

// ---------------------------------------------------------------------------
// Kernel 1: five small GEMMs  U = X @ Wslice   (K = 128), one wave per 16x16
// output tile, V_WMMA_F32_16X16X4_F32 accumulation chain (32 wmma per tile).
//
// Tile map over gridDim.x = 168 blocks of 32 threads:
//   [  0, 32): Ua = h_agv     @ W1d[  0:128]   (4x8 tiles)
//   [ 32, 88): Uo = h_order   @ W1d[128:256]   (7x8 tiles, rows clamped)
//   [ 88,112): Um = h_waiting @ W1d[256:384]   (3x8 tiles)
//   [112,144): Pa = h_agv     @ W1p[  0:128]   (4x8 tiles)
//   [144,168): Pb = h_buffer  @ W1p[128:256]   (3x8 tiles)
// ---------------------------------------------------------------------------
__global__ __launch_bounds__(32) void pn_factor_gemm_wmma(
    const float* __restrict__ hOrder, const float* __restrict__ hAgv,
    const float* __restrict__ hWait,  const float* __restrict__ hBuf,
    const float* __restrict__ W1d,    const float* __restrict__ W1p,
    float* __restrict__ ws) {
  using namespace pn;
  const int bid = blockIdx.x;
  const float* X; const float* W; float* out; int R; int tile;
  if (bid < 32)       { X = hAgv;   W = W1d;            out = ws + OFF_UA; R = 64;  tile = bid; }
  else if (bid < 88)  { X = hOrder; W = W1d + 128 * D;  out = ws + OFF_UO; R = 100; tile = bid - 32; }
  else if (bid < 112) { X = hWait;  W = W1d + 256 * D;  out = ws + OFF_UM; R = 40;  tile = bid - 88; }
  else if (bid < 144) { X = hAgv;   W = W1p;            out = ws + OFF_PA; R = 64;  tile = bid - 112; }
  else                { X = hBuf;   W = W1p + 128 * D;  out = ws + OFF_PB; R = 40;  tile = bid - 144; }

  const int row0 = (tile >> 3) * 16;
  const int col0 = (tile & 7) * 16;
  const int lane = threadIdx.x;
  const int half = lane >> 4;   // K-pair selector (ISA f32 A/B fragment layout)
  const int l16  = lane & 15;

  // A fragment row for this lane (clamped for the 100/40-row edge tiles;
  // duplicated results land only in padded workspace rows, never read back).
  int arow = row0 + l16;
  if (arow > R - 1) arow = R - 1;

#if HAVE_WMMA_F32_16X16X4
  v8f acc = {0.f, 0.f, 0.f, 0.f, 0.f, 0.f, 0.f, 0.f};
  // A: lane l (l<16) holds X[row0+l][k+0..1]; lanes>=16 hold K=k+2..3  (8B-aligned pair)
  const float* aptr = X + arow * D + 2 * half;
  // B: VGPR j holds W[k + 2*half + j][col0 + (lane&15)]
  const float* bptr = W + (2 * half) * D + col0 + l16;
#pragma unroll
  for (int k = 0; k < D; k += 4) {
    v2f a = *(const v2f*)(aptr + k);
    v2f b;
    b.x = bptr[k * D];
    b.y = bptr[(k + 1) * D];
    acc = __builtin_amdgcn_wmma_f32_16x16x4_f32(
        /*neg_a=*/false, a, /*neg_b=*/false, b,
        /*c_mod=*/(short)0, acc, /*reuse_a=*/false, /*reuse_b=*/false);
  }
  // C/D layout: VGPR i -> row row0+i (lanes 0-15) / row0+i+8 (lanes 16-31), col = col0+l16
#pragma unroll
  for (int i = 0; i < 8; ++i) {
    int r = row0 + i + 8 * half;
    out[r * D + col0 + l16] = acc[i];
  }
#else
  // Scalar fallback (same output mapping) — only if the f32 WMMA builtin is absent.
  const int c = col0 + l16;
#pragma unroll
  for (int i = 0; i < 8; ++i) {
    int r = row0 + i + 8 * half;
    int rr = r > R - 1 ? R - 1 : r;
    float s = 0.f;
    for (int k = 0; k < D; ++k) s = fmaf(X[rr * D + k], W[k * D + c], s);
    out[r * D + c] = s;
  }
#endif
}

// ---------------------------------------------------------------------------
// Kernel 2: combine + score + mask. One block per AGV `a`.
//   disp[a,o,m] = dot(relu(Uo[o] + Um[m] + (Ua[a]+b1d)), W2d) + b2d
//   pick[a,m]   = dot(relu(Pb[m] + (Pa[a]+b1p)), W2p) + b2p
// Um/Pb + fused biases staged in LDS (stride 132: conflict-free b128 loads);
// Uo (51 KB) streams through L1/L2.
// ---------------------------------------------------------------------------
__global__ __launch_bounds__(256) void pn_combine_score(
    const float* __restrict__ ws,
    const float* __restrict__ b1d, const float* __restrict__ W2d, const float* __restrict__ b2d,
    const float* __restrict__ b1p, const float* __restrict__ W2p, const float* __restrict__ b2p,
    const int* __restrict__ mask, float* __restrict__ out) {
  using namespace pn;
  __shared__ __align__(16) float sUm[M * LSTR];
  __shared__ __align__(16) float sPb[M * LSTR];
  __shared__ __align__(16) float sBaseD[D];
  __shared__ __align__(16) float sW2dv[D];
  __shared__ __align__(16) float sBaseP[D];
  __shared__ __align__(16) float sW2pv[D];

  const int a = blockIdx.x;
  const int tid = threadIdx.x;
  const float* Ua = ws + OFF_UA + a * D;
  const float* Uo = ws + OFF_UO;
  const float* Um = ws + OFF_UM;
  const float* Pa = ws + OFF_PA + a * D;
  const float* Pb = ws + OFF_PB;

  for (int i = tid; i < M * D; i += 256) {
    int r = i >> 7, h = i & (D - 1);
    sUm[r * LSTR + h] = Um[i];
    sPb[r * LSTR + h] = Pb[i];
  }
  for (int h = tid; h < D; h += 256) {
    sBaseD[h] = Ua[h] + b1d[h];
    sW2dv[h]  = W2d[h];
    sBaseP[h] = Pa[h] + b1p[h];
    sW2pv[h]  = W2p[h];
  }
  __syncthreads();

  const float bias2d = b2d[0];
  const float bias2p = b2p[0];
  const float NEG_INF = -__builtin_inff();

  // Dispatch scores: 4000 per AGV
  for (int p = tid; p < OM; p += 256) {
    const int o = p / M;
    const int m = p - o * M;
    const float* uo = Uo + o * D;
    const float* um = sUm + m * LSTR;
    float s = 0.f;
#pragma unroll 4
    for (int h = 0; h < D; h += 4) {
      float4 u  = *(const float4*)(uo + h);
      float4 v  = *(const float4*)(um + h);
      float4 bb = *(const float4*)(sBaseD + h);
      float4 w  = *(const float4*)(sW2dv + h);
      s = fmaf(fmaxf(u.x + v.x + bb.x, 0.f), w.x, s);
      s = fmaf(fmaxf(u.y + v.y + bb.y, 0.f), w.y, s);
      s = fmaf(fmaxf(u.z + v.z + bb.z, 0.f), w.z, s);
      s = fmaf(fmaxf(u.w + v.w + bb.w, 0.f), w.w, s);
    }
    s += bias2d;
    const int idx = a * LOGITS_PER_A + p;
    out[idx] = (mask[idx] != 0) ? s : NEG_INF;
  }

  // Pickup scores: 40 per AGV
  for (int m = tid; m < M; m += 256) {
    const float* pb = sPb + m * LSTR;
    float s = 0.f;
#pragma unroll 4
    for (int h = 0; h < D; h += 4) {
      float4 v  = *(const float4*)(pb + h);
      float4 bb = *(const float4*)(sBaseP + h);
      float4 w  = *(const float4*)(sW2pv + h);
      s = fmaf(fmaxf(v.x + bb.x, 0.f), w.x, s);
      s = fmaf(fmaxf(v.y + bb.y, 0.f), w.y, s);
      s = fmaf(fmaxf(v.z + bb.z, 0.f), w.z, s);
      s = fmaf(fmaxf(v.w + bb.w, 0.f), w.w, s);
    }
    s += bias2p;
    const int idx = a * LOGITS_PER_A + OM + m;
    out[idx] = (mask[idx] != 0) ? s : NEG_INF;
  }
}

// ---------------------------------------------------------------------------
// Launch. Input order (setup_inputs dict):
//  0 h_order[100*128] 1 h_agv[64*128] 2 h_waiting[40*128] 3 h_buffer[40*128]
//  4 W1d[384*128] 5 b1d[128] 6 W2d[128] 7 b2d[1]
//  8 W1p[256*128] 9 b1p[128] 10 W2p[128] 11 b2p[1] 12 action_mask[258560]
// d_out: 258,560 f32. d_ws needs >= 172,032 bytes.
// ---------------------------------------------------------------------------
extern "C" void kernel_launch(void* const* d_in, const int* in_sizes, int n_in,
                              void* d_out, int out_size, void* d_ws, size_t ws_size,
                              hipStream_t stream) {
  (void)in_sizes; (void)n_in; (void)out_size; (void)ws_size;
  const float* h_order = (const float*)d_in[0];
  const float* h_agv   = (const float*)d_in[1];
  const float* h_wait  = (const float*)d_in[2];
  const float* h_buf   = (const float*)d_in[3];
  const float* W1d     = (const float*)d_in[4];
  const float* b1d     = (const float*)d_in[5];
  const float* W2d     = (const float*)d_in[6];
  const float* b2d     = (const float*)d_in[7];
  const float* W1p     = (const float*)d_in[8];
  const float* b1p     = (const float*)d_in[9];
  const float* W2p     = (const float*)d_in[10];
  const float* b2p     = (const float*)d_in[11];
  const int*   mask    = (const int*)d_in[12];
  float* ws  = (float*)d_ws;
  float* out = (float*)d_out;

  pn_factor_gemm_wmma<<<168, 32, 0, stream>>>(h_order, h_agv, h_wait, h_buf,
                                              W1d, W1p, ws);
  pn_combine_score<<<pn::A, 256, 0, stream>>>(ws, b1d, W2d, b2d,
                                              b1p, W2p, b2p, mask, out);
}